// RGCN_SL_Predictor_16037407883778
// MI455X (gfx1250) — compile-verified
//
#include <hip/hip_runtime.h>
#include <hip/hip_bf16.h>
#include <math.h>

// Problem constants (match reference)
#define N_   20000
#define E_   320000
#define R_   8
#define B_   8
#define IN_  128
#define H_   256
#define OUT_ 128
#define P_   100000
#define LN_EPS 1e-5f

typedef float v8f __attribute__((ext_vector_type(8)));
typedef float v2f __attribute__((ext_vector_type(2)));

#define WMMA_F32(a, b, c) \
  __builtin_amdgcn_wmma_f32_16x16x4_f32(false, (a), false, (b), (short)0, (c), false, false)

// ---------------------------------------------------------------------------
// softmax over relatt (R=8), single thread is fine
// ---------------------------------------------------------------------------
__global__ void relatt_softmax_kernel(const float* __restrict__ relatt,
                                      float* __restrict__ a) {
  if (threadIdx.x == 0 && blockIdx.x == 0) {
    float m = relatt[0];
    for (int r = 1; r < R_; ++r) m = fmaxf(m, relatt[r]);
    float e[R_]; float s = 0.f;
    for (int r = 0; r < R_; ++r) { e[r] = expf(relatt[r] - m); s += e[r]; }
    for (int r = 0; r < R_; ++r) a[r] = e[r] / s;
  }
}

// ---------------------------------------------------------------------------
// Wcat[r,i,o] = a[r] * sum_b att[r,b] * basis[b,i,o]
// ---------------------------------------------------------------------------
__global__ void make_weights_kernel(const float* __restrict__ att,
                                    const float* __restrict__ basis,
                                    const float* __restrict__ a,
                                    float* __restrict__ W,
                                    int in_ch, int out_ch) {
  int idx = blockIdx.x * blockDim.x + threadIdx.x;
  int io_sz = in_ch * out_ch;
  int total = R_ * io_sz;
  if (idx >= total) return;
  int r  = idx / io_sz;
  int io = idx - r * io_sz;
  float s = 0.f;
#pragma unroll
  for (int bb = 0; bb < B_; ++bb)
    s += att[r * B_ + bb] * basis[(size_t)bb * io_sz + io];
  W[idx] = s * a[r];
}

// ---------------------------------------------------------------------------
// agg[(dst*R + et), f] += x[src, f]   — one block per edge, one thread per f
// ---------------------------------------------------------------------------
__global__ void scatter_agg_kernel(const float* __restrict__ x,
                                   const int* __restrict__ src,
                                   const int* __restrict__ dst,
                                   const int* __restrict__ et,
                                   float* __restrict__ agg, int in_ch) {
  int e = blockIdx.x;
  int f = threadIdx.x;
  int s = src[e], d = dst[e], r = et[e];
  atomicAdd(&agg[((size_t)d * R_ + r) * in_ch + f],
            x[(size_t)s * in_ch + f]);
}

// ---------------------------------------------------------------------------
// fp32 WMMA GEMM with 4-way N register blocking:
// C[M,Nc] (+)= A[M,K] @ B[K,Nc], row-major.  One wave computes a 16x64 strip
// (4 accumulators) so each A fragment feeds 4 V_WMMA_F32_16X16X4_F32 issues.
// A fragment: lane (g*16+l) holds A[m=l, k=k0+2g+{0,1}]  (ISA 16x4 layout)
// B fragment: lane (g*16+l) holds B[k=k0+2g+{0,1}, n=l]
// C fragment: vgpr i, lane (g*16+l) -> C[m=i+8g, n=l]
// Requires M%16==0, Nc%64==0, K%16==0 (true for all call sites here).
// ---------------------------------------------------------------------------
__global__ void gemm_f32_wmma_kernel(const float* __restrict__ A,
                                     const float* __restrict__ Bm,
                                     float* __restrict__ C,
                                     int M, int Nc, int K,
                                     int lda, int ldb, int ldc,
                                     int accumulate) {
  int wave = threadIdx.x >> 5;
  int lane = threadIdx.x & 31;
  int l16  = lane & 15;
  int g    = lane >> 4;                 // 0 or 1
  int tilesN = Nc >> 6;                 // 64-column strips
  int tilesTotal = (M >> 4) * tilesN;
  int tile = blockIdx.x * 8 + wave;     // 8 waves per block
  if (tile >= tilesTotal) return;       // wave-uniform exit: EXEC stays full
  int tm = tile / tilesN;
  int tn = tile - tm * tilesN;

  const float* Arow  = A + (size_t)(tm * 16 + l16) * lda + 2 * g;
  const float* Bbase = Bm + (size_t)(2 * g) * ldb + tn * 64 + l16;

  v8f acc0 = {}, acc1 = {}, acc2 = {}, acc3 = {};
  for (int k0 = 0; k0 < K; k0 += 16) {
    __builtin_prefetch(Arow + k0 + 16, 0, 0);   // global_prefetch: next A chunk
#pragma unroll
    for (int kk = 0; kk < 16; kk += 4) {
      int k = k0 + kk;
      v2f a = *(const v2f*)(Arow + k);          // A[m, k+2g .. k+2g+1]
      const float* Bk0 = Bbase + (size_t)k * ldb;
      const float* Bk1 = Bk0 + ldb;
      v2f b0, b1, b2, b3;
      b0.x = Bk0[0];  b0.y = Bk1[0];
      b1.x = Bk0[16]; b1.y = Bk1[16];
      b2.x = Bk0[32]; b2.y = Bk1[32];
      b3.x = Bk0[48]; b3.y = Bk1[48];
      acc0 = WMMA_F32(a, b0, acc0);
      acc1 = WMMA_F32(a, b1, acc1);
      acc2 = WMMA_F32(a, b2, acc2);
      acc3 = WMMA_F32(a, b3, acc3);
    }
  }

  float* Crow = C + (size_t)(tm * 16 + 8 * g) * ldc + tn * 64 + l16;
#pragma unroll
  for (int i = 0; i < 8; ++i) {
    size_t off = (size_t)i * ldc;
    float v0 = acc0[i], v1 = acc1[i], v2 = acc2[i], v3 = acc3[i];
    if (accumulate) {
      v0 += Crow[off];      v1 += Crow[off + 16];
      v2 += Crow[off + 32]; v3 += Crow[off + 48];
    }
    Crow[off]      = v0;
    Crow[off + 16] = v1;
    Crow[off + 32] = v2;
    Crow[off + 48] = v3;
  }
}

// ---------------------------------------------------------------------------
// Decoder layer-0 GEMM with fused pair-feature construction (4-way N block):
//   A[p, k] = (k<128) ? zi+zj : (k<256) ? zi*zj : |zi-zj|   (f = k%128)
//   C[P, 256] = A[P, 384] @ dW0[384, 256]
// The on-the-fly A fragment is reused by 4 WMMA issues.
// ---------------------------------------------------------------------------
__global__ void gemm_pe_wmma_kernel(const float* __restrict__ z,
                                    const int* __restrict__ gi,
                                    const int* __restrict__ gj,
                                    const float* __restrict__ Bm,
                                    float* __restrict__ C,
                                    int M, int Nc, int K,
                                    int ldb, int ldc) {
  int wave = threadIdx.x >> 5;
  int lane = threadIdx.x & 31;
  int l16  = lane & 15;
  int g    = lane >> 4;
  int tilesN = Nc >> 6;
  int tilesTotal = (M >> 4) * tilesN;
  int tile = blockIdx.x * 8 + wave;
  if (tile >= tilesTotal) return;
  int tm = tile / tilesN;
  int tn = tile - tm * tilesN;

  int row = tm * 16 + l16;
  const float* zi = z + (size_t)gi[row] * OUT_;
  const float* zj = z + (size_t)gj[row] * OUT_;
  const float* Bbase = Bm + (size_t)(2 * g) * ldb + tn * 64 + l16;

  v8f acc0 = {}, acc1 = {}, acc2 = {}, acc3 = {};
  for (int k = 0; k < K; k += 4) {
    int kk = k + 2 * g;          // even, so kk and kk+1 share a section
    int f  = kk & (OUT_ - 1);
    int s  = kk >> 7;            // 0: sum, 1: prod, 2: absdiff
    float zix = zi[f], ziy = zi[f + 1];
    float zjx = zj[f], zjy = zj[f + 1];
    v2f a;
    if (s == 0)      { a.x = zix + zjx;        a.y = ziy + zjy; }
    else if (s == 1) { a.x = zix * zjx;        a.y = ziy * zjy; }
    else             { a.x = fabsf(zix - zjx); a.y = fabsf(ziy - zjy); }
    const float* Bk0 = Bbase + (size_t)k * ldb;
    const float* Bk1 = Bk0 + ldb;
    v2f b0, b1, b2, b3;
    b0.x = Bk0[0];  b0.y = Bk1[0];
    b1.x = Bk0[16]; b1.y = Bk1[16];
    b2.x = Bk0[32]; b2.y = Bk1[32];
    b3.x = Bk0[48]; b3.y = Bk1[48];
    acc0 = WMMA_F32(a, b0, acc0);
    acc1 = WMMA_F32(a, b1, acc1);
    acc2 = WMMA_F32(a, b2, acc2);
    acc3 = WMMA_F32(a, b3, acc3);
  }

  float* Crow = C + (size_t)(tm * 16 + 8 * g) * ldc + tn * 64 + l16;
#pragma unroll
  for (int i = 0; i < 8; ++i) {
    size_t off = (size_t)i * ldc;
    Crow[off]      = acc0[i];
    Crow[off + 16] = acc1[i];
    Crow[off + 32] = acc2[i];
    Crow[off + 48] = acc3[i];
  }
}

// ---------------------------------------------------------------------------
// Fused LayerNorm epilogue: out = maybe_gelu( LN(y + bias)*gamma+beta + resid )
// One wave (32 lanes) per row; C in {64,128,256}. In-place safe (out==y).
// ---------------------------------------------------------------------------
__global__ void ln_res_gelu_kernel(const float* __restrict__ y,
                                   const float* __restrict__ bias,
                                   const float* __restrict__ gamma,
                                   const float* __restrict__ beta,
                                   const float* __restrict__ resid,
                                   float* __restrict__ out,
                                   int rows, int C, int do_gelu) {
  int wave = threadIdx.x >> 5;
  int lane = threadIdx.x & 31;
  int row = blockIdx.x * 8 + wave;
  if (row >= rows) return;
  const float* yr = y + (size_t)row * C;
  int nv = C >> 5;                       // elements per lane (<=8)
  float vals[8];
  float s = 0.f;
  for (int i = 0; i < nv; ++i) {
    int c = lane + (i << 5);
    float v = yr[c];
    if (bias) v += bias[c];
    vals[i] = v;
    s += v;
  }
  for (int o = 16; o; o >>= 1) s += __shfl_xor(s, o, 32);
  float mean = s / (float)C;
  float vs = 0.f;
  for (int i = 0; i < nv; ++i) { float d = vals[i] - mean; vs += d * d; }
  for (int o = 16; o; o >>= 1) vs += __shfl_xor(vs, o, 32);
  float inv = rsqrtf(vs / (float)C + LN_EPS);
  for (int i = 0; i < nv; ++i) {
    int c = lane + (i << 5);
    float v = (vals[i] - mean) * inv * gamma[c] + beta[c];
    if (resid) v += resid[(size_t)row * C + c];
    if (do_gelu) v = 0.5f * v * (1.f + erff(v * 0.70710678118654752f));
    out[(size_t)row * C + c] = v;
  }
}

// ---------------------------------------------------------------------------
// logits[p] = dot(d2[p,0:64], w[0:64]) + b    — one wave per pair
// ---------------------------------------------------------------------------
__global__ void dot64_kernel(const float* __restrict__ d2,
                             const float* __restrict__ w,
                             const float* __restrict__ b,
                             float* __restrict__ logits, int rows) {
  int wave = threadIdx.x >> 5;
  int lane = threadIdx.x & 31;
  int p = blockIdx.x * 8 + wave;
  if (p >= rows) return;
  const float* dr = d2 + (size_t)p * 64;
  float s = dr[lane] * w[lane] + dr[lane + 32] * w[lane + 32];
  for (int o = 16; o; o >>= 1) s += __shfl_xor(s, o, 32);
  if (lane == 0) logits[p] = s + b[0];
}

// ---------------------------------------------------------------------------
// host-side orchestration
// ---------------------------------------------------------------------------
extern "C" void kernel_launch(void* const* d_in, const int* in_sizes, int n_in,
                              void* d_out, int out_size, void* d_ws, size_t ws_size,
                              hipStream_t stream) {
  (void)in_sizes; (void)n_in; (void)out_size; (void)ws_size;

  const float* x    = (const float*)d_in[0];
  const int*   ei   = (const int*)d_in[1];
  const int*   esrc = ei;
  const int*   edst = ei + E_;
  const int*   et   = (const int*)d_in[2];
  const int*   gp   = (const int*)d_in[3];
  const int*   gi   = gp;
  const int*   gj   = gp + P_;

  const float* basis[4], *att[4], *root[4], *relatt[4], *gam[4], *bet[4];
  for (int i = 0; i < 4; ++i) {
    basis[i]  = (const float*)d_in[4 + 6 * i + 0];
    att[i]    = (const float*)d_in[4 + 6 * i + 1];
    root[i]   = (const float*)d_in[4 + 6 * i + 2];
    relatt[i] = (const float*)d_in[4 + 6 * i + 3];
    gam[i]    = (const float*)d_in[4 + 6 * i + 4];
    bet[i]    = (const float*)d_in[4 + 6 * i + 5];
  }
  const float* res0 = (const float*)d_in[28];
  const float* res3 = (const float*)d_in[29];
  const float* dW0 = (const float*)d_in[30], *db0 = (const float*)d_in[31];
  const float* dg0 = (const float*)d_in[32], *dbb0 = (const float*)d_in[33];
  const float* dW1 = (const float*)d_in[34], *db1 = (const float*)d_in[35];
  const float* dg1 = (const float*)d_in[36], *dbb1 = (const float*)d_in[37];
  const float* dW2 = (const float*)d_in[38], *db2 = (const float*)d_in[39];
  const float* dg2 = (const float*)d_in[40], *dbb2 = (const float*)d_in[41];
  const float* dW3 = (const float*)d_in[42], *db3 = (const float*)d_in[43];

  // workspace layout (floats)
  float* ws = (float*)d_ws;
  size_t off = 0;
  float* a_sm  = ws + off; off += 64;                       // softmax(relatt)
  float* Wcat  = ws + off; off += (size_t)R_ * H_ * H_;     // 524288
  float* hA    = ws + off; off += (size_t)N_ * H_;          // 5.12M
  float* hB    = ws + off; off += (size_t)N_ * H_;          // 5.12M (contiguous after hA)
  float* resid = ws + off; off += (size_t)N_ * H_;          // 5.12M
  float* big   = ws + off;                                  // 40.96M: agg, later d0+d1
  float* d0 = big;
  float* d1 = big + (size_t)P_ * H_;
  float* d2 = hA;   // reuses hA+hB region (free after z copied out)

  auto gemm = [&](const float* A, const float* Bm, float* Cm,
                  int M, int Nc, int K, int lda, int ldb, int ldc, int acc) {
    int tiles = (M / 16) * (Nc / 64);
    gemm_f32_wmma_kernel<<<(tiles + 7) / 8, 256, 0, stream>>>(
        A, Bm, Cm, M, Nc, K, lda, ldb, ldc, acc);
  };

  // ---- RGCN layers -------------------------------------------------------
  // layer i: cur(in) -> conv buffer(out); resid = resmat? cur@resmat : cur
  struct { const float* cur; int in; int out; int li;
           const float* resmat; float* conv; int gelu; } L[4] = {
    { x,  IN_, H_,   0, res0,    hA, 1 },
    { hA, H_,  H_,   1, nullptr, hB, 1 },
    { hB, H_,  H_,   2, nullptr, hA, 1 },
    { hA, H_,  OUT_, 3, res3,    hB, 0 },
  };

  for (int i = 0; i < 4; ++i) {
    const float* cur = L[i].cur;
    int in_ch = L[i].in, out_ch = L[i].out, li = L[i].li;
    float* conv = L[i].conv;

    relatt_softmax_kernel<<<1, 32, 0, stream>>>(relatt[li], a_sm);
    int wtotal = R_ * in_ch * out_ch;
    make_weights_kernel<<<(wtotal + 255) / 256, 256, 0, stream>>>(
        att[li], basis[li], a_sm, Wcat, in_ch, out_ch);

    hipMemsetAsync(big, 0, (size_t)N_ * R_ * in_ch * sizeof(float), stream);
    scatter_agg_kernel<<<E_, in_ch, 0, stream>>>(cur, esrc, edst, et, big, in_ch);

    // conv = agg @ Wcat ; conv += cur @ root
    gemm(big, Wcat, conv, N_, out_ch, R_ * in_ch, R_ * in_ch, out_ch, out_ch, 0);
    gemm(cur, root[li], conv, N_, out_ch, in_ch, in_ch, out_ch, out_ch, 1);

    const float* rptr;
    if (L[i].resmat) {
      gemm(cur, L[i].resmat, resid, N_, out_ch, in_ch, in_ch, out_ch, out_ch, 0);
      rptr = resid;
    } else {
      rptr = cur;
    }
    ln_res_gelu_kernel<<<(N_ + 7) / 8, 256, 0, stream>>>(
        conv, nullptr, gam[li], bet[li], rptr, conv, N_, out_ch, L[i].gelu);
  }
  // z lives in hB

  // emit z into d_out (after logits region)
  hipMemcpyAsync((float*)d_out + P_, hB, (size_t)N_ * OUT_ * sizeof(float),
                 hipMemcpyDeviceToDevice, stream);

  // ---- decoder -----------------------------------------------------------
  {
    int tiles = (P_ / 16) * (H_ / 64);
    gemm_pe_wmma_kernel<<<(tiles + 7) / 8, 256, 0, stream>>>(
        hB, gi, gj, dW0, d0, P_, H_, 3 * OUT_, H_, H_);
  }
  ln_res_gelu_kernel<<<(P_ + 7) / 8, 256, 0, stream>>>(
      d0, db0, dg0, dbb0, nullptr, d0, P_, H_, 1);

  gemm(d0, dW1, d1, P_, H_ / 2, H_, H_, H_ / 2, H_ / 2, 0);
  ln_res_gelu_kernel<<<(P_ + 7) / 8, 256, 0, stream>>>(
      d1, db1, dg1, dbb1, nullptr, d1, P_, H_ / 2, 1);

  gemm(d1, dW2, d2, P_, H_ / 4, H_ / 2, H_ / 2, H_ / 4, H_ / 4, 0);
  ln_res_gelu_kernel<<<(P_ + 7) / 8, 256, 0, stream>>>(
      d2, db2, dg2, dbb2, nullptr, d2, P_, H_ / 4, 1);

  dot64_kernel<<<(P_ + 7) / 8, 256, 0, stream>>>(d2, dW3, db3, (float*)d_out, P_);
}